// ResidualBlock_4037269258944
// MI455X (gfx1250) — compile-verified
//
#include <hip/hip_runtime.h>

typedef __attribute__((ext_vector_type(2))) float v2f;
typedef __attribute__((ext_vector_type(8))) float v8f;

#define NVOX 500000
#define CCH  32
#define K27  27
#define WAVES_PER_BLOCK 16
#define BLOCK (WAVES_PER_BLOCK * 32)
#define NTILES (NVOX / 16)                 // 31250, exact
#define LDS_BYTES (K27 * CCH * CCH * 4)    // 110592 B, fits 320KB WGP LDS

static __device__ __forceinline__ v8f wmma4(v2f a, v2f b, v8f c) {
    // D = A(16x4,f32) x B(4x16,f32) + C(16x16,f32)
    return __builtin_amdgcn_wmma_f32_16x16x4_f32(
        /*neg_a=*/false, a, /*neg_b=*/false, b,
        /*c_mod=*/(short)0, c, /*reuse_a=*/false, /*reuse_b=*/false);
}

// FIRST=true : dst = relu(conv(src=feats, W))          (dst has NVOX+1 rows, row N zeroed)
// FIRST=false: dst = relu(conv(src=h,     W) + feats)
template <bool FIRST>
__global__ __launch_bounds__(BLOCK)
void sconv_kernel(const float* __restrict__ src,
                  const float* __restrict__ feats,
                  const int*   __restrict__ nbr,   // [27, NVOX]
                  const float* __restrict__ W,     // [27, 32, 32]
                  float*       __restrict__ dst)
{
    // Pair-interleaved weight stage: sWp[((k*8+kk)*2+half)*32 + col] =
    //   { W[k][4kk+2half][col], W[k][4kk+2half+1][col] }
    // so each lane's WMMA B operand is one contiguous 8B LDS word.
    extern __shared__ float sW[];
    v2f* sWp = (v2f*)sW;
    for (int i = threadIdx.x; i < K27 * 8 * 2 * CCH; i += BLOCK) {
        const int col = i & 31;
        int t = i >> 5;
        const int h  = t & 1;  t >>= 1;
        const int kk = t & 7;
        const int k  = t >> 3;
        const int r  = 4 * kk + 2 * h;
        v2f p;
        p.x = W[((size_t)k * CCH + r    ) * CCH + col];
        p.y = W[((size_t)k * CCH + r + 1) * CCH + col];
        sWp[i] = p;
    }
    __syncthreads();

    if (FIRST && blockIdx.x == 0 && threadIdx.x < CCH)
        dst[(size_t)NVOX * CCH + threadIdx.x] = 0.0f;   // sentinel zero row of h

    const int wave = threadIdx.x >> 5;
    const int lane = threadIdx.x & 31;
    const int tile = blockIdx.x * WAVES_PER_BLOCK + wave;
    if (tile >= NTILES) return;                    // uniform per-wave tail guard

    const int m    = lane & 15;                    // voxel-in-tile / N-in-half
    const int half = lane >> 4;
    const int base = tile * 16;

    v8f c_lo = {};                                 // out channels 0..15
    v8f c_hi = {};                                 // out channels 16..31

    int idxc = nbr[base + m];                      // k=0 indices (prefetched)

    #pragma unroll 1
    for (int k = 0; k < K27; ++k) {
        const int idxn = (k < K27 - 1) ? nbr[(size_t)(k + 1) * NVOX + base + m] : 0;

        // ---- gather A tile: lane (m,half) holds channels 4*kk+2*half .. +1 ----
        v2f a[8];
        if (FIRST) {
            const bool valid = (idxc < NVOX);
            const float* rp = src + (size_t)(valid ? idxc : 0) * CCH + 2 * half;
            #pragma unroll
            for (int kk = 0; kk < 8; ++kk) {
                v2f t = *(const v2f*)(rp + 4 * kk);
                a[kk].x = valid ? t.x : 0.0f;
                a[kk].y = valid ? t.y : 0.0f;
            }
        } else {
            const float* rp = src + (size_t)idxc * CCH + 2 * half;  // row NVOX is zeros
            #pragma unroll
            for (int kk = 0; kk < 8; ++kk)
                a[kk] = *(const v2f*)(rp + 4 * kk);
        }

        // ---- 8 K-chunks of 4; B pairs load directly from LDS, no repacking ----
        const v2f* wk = sWp + (size_t)k * 8 * 2 * CCH;
        #pragma unroll
        for (int kk = 0; kk < 8; ++kk) {
            const v2f* wc = wk + (kk * 2 + half) * CCH;
            const v2f b_lo = wc[m];            // B VGPR0/1 = K{2h, 2h+1}, N = m
            const v2f b_hi = wc[16 + m];       // same rows, out channels 16..31
            c_lo = wmma4(a[kk], b_lo, c_lo);
            c_hi = wmma4(a[kk], b_hi, c_hi);
        }
        idxc = idxn;
    }

    // ---- epilogue: C/D layout (reg v, lane) -> row base+v+8*half, col m / 16+m ----
    #pragma unroll
    for (int v = 0; v < 8; ++v) {
        const int row = base + v + 8 * half;
        float xlo = c_lo[v];
        float xhi = c_hi[v];
        if (!FIRST) {
            xlo += feats[(size_t)row * CCH + m];
            xhi += feats[(size_t)row * CCH + 16 + m];
        }
        dst[(size_t)row * CCH + m]      = fmaxf(xlo, 0.0f);
        dst[(size_t)row * CCH + 16 + m] = fmaxf(xhi, 0.0f);
    }
}

extern "C" void kernel_launch(void* const* d_in, const int* in_sizes, int n_in,
                              void* d_out, int out_size, void* d_ws, size_t ws_size,
                              hipStream_t stream) {
    const float* feats = (const float*)d_in[0];   // [N,32] f32
    const int*   nbr   = (const int*)  d_in[1];   // [27,N] i32
    const float* W1    = (const float*)d_in[2];   // [27,32,32] f32
    const float* W2    = (const float*)d_in[3];   // [27,32,32] f32
    float* out = (float*)d_out;                   // [N,32] f32
    float* h   = (float*)d_ws;                    // [(N+1),32] f32 scratch (zero row at N)

    const int nblocks = (NTILES + WAVES_PER_BLOCK - 1) / WAVES_PER_BLOCK;

    sconv_kernel<true ><<<nblocks, BLOCK, LDS_BYTES, stream>>>(feats, nullptr, nbr, W1, h);
    sconv_kernel<false><<<nblocks, BLOCK, LDS_BYTES, stream>>>(h,     feats,   nbr, W2, out);
}